// AttributeMemoryFusion_27419071218472
// MI455X (gfx1250) — compile-verified
//
#include <hip/hip_runtime.h>
#include <hip/hip_bf16.h>
#include <math.h>

// Problem constants (from reference): B=8192, M=64, D=256
#define BATCH 8192
#define MEMN  64
#define DIM   256
#define NEGV  (-1e9f)

// Padded LDS row strides (floats) to avoid bank conflicts (64 banks of 4B)
#define MPAD  260   // 260 % 64 == 4  -> dot phase 4-way max, col phase conflict-free
#define HPAD  260

typedef __attribute__((ext_vector_type(2))) float v2f;
typedef __attribute__((ext_vector_type(8))) float v8f;

// Types for the async global->LDS builtin:
//   param0: 'int __vector(4) __device__ *'  (non-const AS1 pointer to int4)
//   param1: 'int __vector(4) __shared__ *'  (AS3 pointer to int4)
typedef int vint4 __attribute__((vector_size(16)));
#define ASYNC_GPTR(p) ((__attribute__((address_space(1))) vint4*)(p))
#define ASYNC_LPTR(p) ((__attribute__((address_space(3))) vint4*)(p))

// CDNA5 async global->LDS staging path (bypasses VGPRs, tracked by ASYNCcnt)
#if defined(__has_builtin)
#if __has_builtin(__builtin_amdgcn_global_load_async_to_lds_b128) && \
    __has_builtin(__builtin_amdgcn_s_wait_asynccnt)
#define USE_ASYNC_LDS 1
#endif
#if __has_builtin(__builtin_amdgcn_rcpf)
#define USE_FAST_RCP 1
#endif
#endif

__device__ __forceinline__ float fast_rcp(float x) {
#ifdef USE_FAST_RCP
    return __builtin_amdgcn_rcpf(x);   // v_rcp_f32, 1 ULP
#else
    return 1.0f / x;
#endif
}

__device__ __forceinline__ float fast_sigmoid(float x) {
    return fast_rcp(1.0f + __expf(-x));
}

// -----------------------------------------------------------------------------
// Kernel A: per-sample masked attention over ragged memory.
// One workgroup (256 threads = 8 wave32) per sample. mem[b] (64x256 fp32, 64KB)
// is streamed from HBM exactly once into LDS via GLOBAL_LOAD_ASYNC_TO_LDS_B128
// (no VGPR round-trip, tracked by ASYNCcnt); scores, masked softmax, and the
// attention-weighted sum r are computed out of LDS. r -> workspace.
// -----------------------------------------------------------------------------
__global__ __launch_bounds__(256) void attn_reduce_kernel(
    const float* __restrict__ h_tilde,   // [B, D]
    const float* __restrict__ mem,       // [B, M, D]
    const int*   __restrict__ lengths,   // [B]
    float*       __restrict__ r_out)     // [B, D]  (workspace)
{
    __shared__ float mem_s[MEMN * MPAD]; // 65 KB, padded rows
    __shared__ float h_s[DIM];
    __shared__ float part_s[256];        // 4 partial sums x 64 rows
    __shared__ float attn_s[MEMN];

    const int b   = blockIdx.x;
    const int tid = threadIdx.x;
    const float* __restrict__ memb = mem + (size_t)b * MEMN * DIM;

    // ---- Stage mem[b] into LDS: 4096 float4 chunks / 256 threads = 16 each ----
#ifdef USE_ASYNC_LDS
#pragma unroll
    for (int i = 0; i < 16; ++i) {
        const int c    = tid + i * 256;       // chunk id in [0, 4096)
        const int mrow = c >> 6;              // / 64 chunks per row
        const int col  = (c & 63) << 2;       // *4 floats
        __builtin_amdgcn_global_load_async_to_lds_b128(
            ASYNC_GPTR(memb + mrow * DIM + col),
            ASYNC_LPTR(mem_s + mrow * MPAD + col),
            /*imm offset*/ 0, /*cpol*/ 0);
    }
    h_s[tid] = h_tilde[(size_t)b * DIM + tid];
    __builtin_amdgcn_s_wait_asynccnt(0);      // this wave's async LDS writes done
#else
#pragma unroll
    for (int i = 0; i < 16; ++i) {
        const int c    = tid + i * 256;
        const int mrow = c >> 6;
        const int col  = (c & 63) << 2;
        float4 v = *(const float4*)(memb + mrow * DIM + col);
        *(float4*)(mem_s + mrow * MPAD + col) = v;
    }
    h_s[tid] = h_tilde[(size_t)b * DIM + tid];
#endif
    __syncthreads();

    // ---- Scores: 4 threads per row, each covers 64 of the 256 dims ----
    {
        const int mrow = tid & 63;
        const int seg  = tid >> 6;            // 0..3
        const float* __restrict__ row = mem_s + mrow * MPAD + seg * 64;
        const float* __restrict__ hh  = h_s + seg * 64;
        float acc = 0.0f;
#pragma unroll 8
        for (int d = 0; d < 64; ++d) acc = fmaf(row[d], hh[d], acc);
        part_s[seg * 64 + mrow] = acc;
    }
    __syncthreads();

    // ---- Masked softmax over 64 slots, done by wave 0 (2 rows per lane) ----
    const int len = lengths[b];
    if (tid < 32) {
        float s0 = part_s[tid]      + part_s[64 + tid]  + part_s[128 + tid] + part_s[192 + tid];
        float s1 = part_s[32 + tid] + part_s[96 + tid]  + part_s[160 + tid] + part_s[224 + tid];
        if (tid      >= len) s0 = NEGV;
        if (tid + 32 >= len) s1 = NEGV;
        float mx = fmaxf(s0, s1);
#pragma unroll
        for (int off = 16; off >= 1; off >>= 1)
            mx = fmaxf(mx, __shfl_xor(mx, off, 32));
        const float e0 = __expf(s0 - mx);
        const float e1 = __expf(s1 - mx);
        float sum = e0 + e1;
#pragma unroll
        for (int off = 16; off >= 1; off >>= 1)
            sum += __shfl_xor(sum, off, 32);
        const float inv = fast_rcp(sum);
        attn_s[tid]      = e0 * inv;
        attn_s[tid + 32] = e1 * inv;
    }
    __syncthreads();

    // ---- r[d] = sum_m attn[m] * mem[m][d]; d = tid (column reads: conflict-free)
    float acc = 0.0f;
#pragma unroll 8
    for (int mm = 0; mm < MEMN; ++mm)
        acc = fmaf(attn_s[mm], mem_s[mm * MPAD + tid], acc);
    r_out[(size_t)b * DIM + tid] = acc;
}

// -----------------------------------------------------------------------------
// Kernel B: gated fusion as a batched GEMM via V_WMMA_F32_16X16X4_F32.
//   S = H @ Wg^T + R @ Ug^T + (Wg_b + Ug_b + b_g);  g = sigmoid(S)
//   out = has_mem ? g*r + (1-g)*h : h
// One workgroup per 16 samples; H/R rows staged in LDS (padded stride 260).
// Each wave32 owns a 16(batch) x 16(feature) C tile; K-loop 256/4 = 64 steps,
// two WMMAs per step accumulated into a single v8f C. 8 waves x 2 tiles = 16
// feature tiles = full 256-wide output row block.
// -----------------------------------------------------------------------------
__global__ __launch_bounds__(256) void gate_fuse_kernel(
    const float* __restrict__ h_tilde,   // [B, D]
    const float* __restrict__ r_in,      // [B, D]
    const int*   __restrict__ lengths,   // [B]
    const float* __restrict__ Wg_w,      // [D, D] row-major (j, d)
    const float* __restrict__ Wg_b,      // [D]
    const float* __restrict__ Ug_w,      // [D, D]
    const float* __restrict__ Ug_b,      // [D]
    const float* __restrict__ b_g,       // [D]
    float*       __restrict__ out)       // [B, D]
{
    __shared__ float h_s[16 * HPAD];
    __shared__ float r_s[16 * HPAD];
    __shared__ float bias_s[DIM];
    __shared__ int   len_s[16];

    const int tid   = threadIdx.x;
    const int bBase = blockIdx.x * 16;

    // ---- Stage H and R row blocks (16 x 256 each): 1024 float4 chunks apiece
#ifdef USE_ASYNC_LDS
#pragma unroll
    for (int i = 0; i < 4; ++i) {
        const int c    = tid + i * 256;       // [0, 1024)
        const int mrow = c >> 6;
        const int col  = (c & 63) << 2;
        __builtin_amdgcn_global_load_async_to_lds_b128(
            ASYNC_GPTR(h_tilde + (size_t)(bBase + mrow) * DIM + col),
            ASYNC_LPTR(h_s + mrow * HPAD + col), 0, 0);
        __builtin_amdgcn_global_load_async_to_lds_b128(
            ASYNC_GPTR(r_in + (size_t)(bBase + mrow) * DIM + col),
            ASYNC_LPTR(r_s + mrow * HPAD + col), 0, 0);
    }
    bias_s[tid] = Wg_b[tid] + Ug_b[tid] + b_g[tid];
    if (tid < 16) len_s[tid] = lengths[bBase + tid];
    __builtin_amdgcn_s_wait_asynccnt(0);
#else
#pragma unroll
    for (int i = 0; i < 4; ++i) {
        const int c    = tid + i * 256;
        const int mrow = c >> 6;
        const int col  = (c & 63) << 2;
        float4 hv = *(const float4*)(h_tilde + (size_t)(bBase + mrow) * DIM + col);
        float4 rv = *(const float4*)(r_in    + (size_t)(bBase + mrow) * DIM + col);
        *(float4*)(h_s + mrow * HPAD + col) = hv;
        *(float4*)(r_s + mrow * HPAD + col) = rv;
    }
    bias_s[tid] = Wg_b[tid] + Ug_b[tid] + b_g[tid];
    if (tid < 16) len_s[tid] = lengths[bBase + tid];
#endif
    __syncthreads();

    const int wave = tid >> 5;
    const int lane = tid & 31;
    const int n    = lane & 15;               // A: M=lane&15 ; B: N=lane&15
    const int kk   = (lane >> 4) << 1;        // lanes 16-31 carry K+2,K+3

#if defined(__has_builtin) && __has_builtin(__builtin_amdgcn_wmma_f32_16x16x4_f32)
    for (int t = 0; t < 2; ++t) {
        const int jBase = (wave + t * 8) * 16;
        const float* __restrict__ wg = Wg_w + (size_t)(jBase + n) * DIM + kk;
        const float* __restrict__ ug = Ug_w + (size_t)(jBase + n) * DIM + kk;
        const float* __restrict__ hs = h_s + n * HPAD + kk;   // A row m == lane&15
        const float* __restrict__ rs = r_s + n * HPAD + kk;

        v8f c = {};
#pragma unroll 4
        for (int k0 = 0; k0 < DIM; k0 += 4) {
            v2f aH = *(const v2f*)(hs + k0);
            v2f aR = *(const v2f*)(rs + k0);
            v2f bW = *(const v2f*)(wg + k0);
            v2f bU = *(const v2f*)(ug + k0);
            // D = A*B + C, fp32 exact:  v_wmma_f32_16x16x4_f32
            c = __builtin_amdgcn_wmma_f32_16x16x4_f32(
                    false, aH, false, bW, (short)0, c, false, false);
            c = __builtin_amdgcn_wmma_f32_16x16x4_f32(
                    false, aR, false, bU, (short)0, c, false, false);
        }

        // C layout: vgpr i -> M=i (lanes 0-15) / M=8+i (lanes 16-31), N=lane&15
        const int mOff = (lane >> 4) * 8;
#pragma unroll
        for (int i = 0; i < 8; ++i) {
            const int mm = mOff + i;
            const int j  = jBase + n;
            const float g  = fast_sigmoid(c[i] + bias_s[j]);
            const float rv = r_s[mm * HPAD + j];
            const float hv = h_s[mm * HPAD + j];
            const float fused = (len_s[mm] > 0) ? fmaf(g, rv - hv, hv) : hv;
            out[(size_t)(bBase + mm) * DIM + j] = fused;
        }
    }
#else
    // Scalar fallback (compiles even if the WMMA builtin is absent)
    (void)wave; (void)lane; (void)n; (void)kk;
#pragma unroll
    for (int i = 0; i < 16; ++i) {
        const int idx = tid + i * 256;        // [0, 4096)
        const int mm  = idx >> 8;
        const int j   = idx & 255;
        float acc = bias_s[j];
        const float* __restrict__ wg = Wg_w + (size_t)j * DIM;
        const float* __restrict__ ug = Ug_w + (size_t)j * DIM;
        for (int d = 0; d < DIM; ++d)
            acc += h_s[mm * HPAD + d] * wg[d] + r_s[mm * HPAD + d] * ug[d];
        const float g  = fast_sigmoid(acc);
        const float rv = r_s[mm * HPAD + j];
        const float hv = h_s[mm * HPAD + j];
        const float fused = (len_s[mm] > 0) ? fmaf(g, rv - hv, hv) : hv;
        out[(size_t)(bBase + mm) * DIM + j] = fused;
    }
#endif
}

// -----------------------------------------------------------------------------
extern "C" void kernel_launch(void* const* d_in, const int* in_sizes, int n_in,
                              void* d_out, int out_size, void* d_ws, size_t ws_size,
                              hipStream_t stream) {
    (void)in_sizes; (void)n_in; (void)out_size; (void)ws_size;
    const float* h_tilde = (const float*)d_in[0];
    const float* mem     = (const float*)d_in[1];
    const int*   lengths = (const int*)d_in[2];
    const float* Wg_w    = (const float*)d_in[3];
    const float* Wg_b    = (const float*)d_in[4];
    const float* Ug_w    = (const float*)d_in[5];
    const float* Ug_b    = (const float*)d_in[6];
    const float* b_g     = (const float*)d_in[7];
    float* out  = (float*)d_out;
    float* r_ws = (float*)d_ws;   // B*D floats = 8 MB scratch for r

    attn_reduce_kernel<<<BATCH, 256, 0, stream>>>(h_tilde, mem, lengths, r_ws);
    gate_fuse_kernel<<<BATCH / 16, 256, 0, stream>>>(h_tilde, r_ws, lengths,
                                                     Wg_w, Wg_b, Ug_w, Ug_b, b_g, out);
}